// MultiHeadAdditiveAttention_12043088298062
// MI455X (gfx1250) — compile-verified
//
#include <hip/hip_runtime.h>
#include <hip/hip_bf16.h>

typedef __attribute__((ext_vector_type(2))) float v2f;
typedef __attribute__((ext_vector_type(4))) float v4f;
typedef __attribute__((ext_vector_type(8))) float v8f;
typedef __attribute__((ext_vector_type(4))) int   v4i;

constexpr int BSZ    = 8;
constexpr int NHEAD  = 16;
constexpr int N      = 1024;
constexpr int HIDDEN = 64;

// ---------------------------------------------------------------------------
// Phase 1: sq[b,h,n] = sum_d Q[b,h,n,d] * W[d]        (sel = 0)
//          sk[b,h,n] = sum_d K[b,h,n,d] * W[64 + d]   (sel = 1)
// One wave per 16-row tile, reduction via V_WMMA_F32_16X16X4_F32:
//   D[i,j] = sum_k A[i,k] * B[k,j],  B[k,j] = Wseg[k]  =>  D[i,*] = sq[i]
// 32-bit A 16x4 layout: lanes 0-15 -> K=0,1 in vgpr0,1 ; lanes 16-31 -> K=2,3.
// B (4x16) mirrors the same per-lane K assignment.
// ---------------------------------------------------------------------------
__global__ __launch_bounds__(256)
void mhaa_proj_wmma(const float* __restrict__ q,
                    const float* __restrict__ k,
                    const float* __restrict__ W,
                    float* __restrict__ sq,
                    float* __restrict__ sk) {
    const int lane = threadIdx.x & 31;
    const int wave = blockIdx.x * (blockDim.x >> 5) + (threadIdx.x >> 5);

    constexpr int tilesPerMat = BSZ * NHEAD * (N / 16);   // 8192
    const int sel = wave / tilesPerMat;                   // 0 = sq, 1 = sk
    const int t   = wave % tilesPerMat;
    const int bh  = t / (N / 16);
    const int n0  = (t % (N / 16)) * 16;

    const float* __restrict__ src = sel ? k : q;
    const float* __restrict__ w   = W + sel * HIDDEN;
    float*       __restrict__ dst = sel ? sk : sq;

    const int row  = n0 + (lane & 15);
    const int dsel = (lane >> 4) << 1;                    // 0 or 2
    const float* __restrict__ rowp = src + ((size_t)bh * N + row) * HIDDEN;

    v8f c = {};
    #pragma unroll
    for (int kb = 0; kb < HIDDEN; kb += 4) {
        v2f a, b;
        a.x = rowp[kb + dsel];
        a.y = rowp[kb + dsel + 1];
        b.x = w[kb + dsel];
        b.y = w[kb + dsel + 1];
        // 8 args: (neg_a, A, neg_b, B, c_mod, C, reuse_a, reuse_b)
        c = __builtin_amdgcn_wmma_f32_16x16x4_f32(false, a, false, b,
                                                  (short)0, c, false, false);
    }

    // D layout (16x16 f32): vgpr v, lanes 0-15 -> M=v; lanes 16-31 -> M=v+8.
    // Column N=0 lives in lane 0 (rows 0-7) and lane 16 (rows 8-15).
    float* __restrict__ out = dst + (size_t)bh * N + n0;
    if (lane == 0) {
        #pragma unroll
        for (int v = 0; v < 8; ++v) out[v] = c[v];
    } else if (lane == 16) {
        #pragma unroll
        for (int v = 0; v < 8; ++v) out[8 + v] = c[v];
    }
}

// ---------------------------------------------------------------------------
// Phase 2 (bandwidth-bound): out[b,h,i,j] = mask[b,i,j] ? sq[b,h,i]+sk[b,h,j] : 0
// One block per (b, i) row; thread owns 4 contiguous j. Mask loaded once
// (non-temporal, single-use stream) and reused in registers across all 16
// heads; output written with non-temporal 128-bit stores to keep the 512 MiB
// stream out of L2 (sq + sk stay resident).
// ---------------------------------------------------------------------------
__global__ __launch_bounds__(256)
void mhaa_broadcast_mask(const int* __restrict__ mask,
                         const float* __restrict__ sq,
                         const float* __restrict__ sk,
                         float* __restrict__ out) {
    const int b = blockIdx.x / N;
    const int i = blockIdx.x % N;
    const int j = threadIdx.x << 2;                       // 0..1020, step 4

    const v4i m4 = __builtin_nontemporal_load(
        reinterpret_cast<const v4i*>(mask + ((size_t)b * N + i) * N + j));
    const bool m0 = m4.x > 0, m1 = m4.y > 0, m2 = m4.z > 0, m3 = m4.w > 0;

    #pragma unroll
    for (int h = 0; h < NHEAD; ++h) {
        const size_t bh  = (size_t)b * NHEAD + h;
        const float  sqv = sq[bh * N + i];                // wave-uniform
        const v4f    skv = *reinterpret_cast<const v4f*>(sk + bh * N + j);

        v4f o;
        o.x = m0 ? (sqv + skv.x) : 0.0f;
        o.y = m1 ? (sqv + skv.y) : 0.0f;
        o.z = m2 ? (sqv + skv.z) : 0.0f;
        o.w = m3 ? (sqv + skv.w) : 0.0f;

        __builtin_nontemporal_store(
            o, reinterpret_cast<v4f*>(out + (bh * N + i) * N + j));
    }
}

extern "C" void kernel_launch(void* const* d_in, const int* in_sizes, int n_in,
                              void* d_out, int out_size, void* d_ws, size_t ws_size,
                              hipStream_t stream) {
    const float* q    = (const float*)d_in[0];   // (8,16,1024,64) f32
    const float* kk   = (const float*)d_in[1];   // (8,16,1024,64) f32
    const int*   mask = (const int*)d_in[2];     // (8,1024,1024) i32
    const float* W    = (const float*)d_in[3];   // (1,128) f32
    float*       out  = (float*)d_out;           // (8,16,1024,1024) f32

    float* sq = (float*)d_ws;                    // BSZ*NHEAD*N floats
    float* sk = sq + (size_t)BSZ * NHEAD * N;    // BSZ*NHEAD*N floats (1 MiB total)

    // Phase 1: 2 * 8192 wave-tiles, 8 waves per 256-thread block -> 2048 blocks.
    mhaa_proj_wmma<<<2048, 256, 0, stream>>>(q, kk, W, sq, sk);

    // Phase 2: one block per (b, i) row.
    mhaa_broadcast_mask<<<BSZ * N, 256, 0, stream>>>(mask, sq, sk, out);
}